// RecurrentGraphNeuralNet_39496519254124
// MI455X (gfx1250) — compile-verified
//
#include <hip/hip_runtime.h>

#define N_NODES 100000
#define N_EDGES 1600000
#define HID 128
#define NODE_CH 128
#define PRED 40

typedef float v2f __attribute__((ext_vector_type(2)));
typedef float v8f __attribute__((ext_vector_type(8)));

// ---------------------------------------------------------------------------
// Kernel 1: zero the scatter workspace (agg[N][HID] + deg[N])
// ---------------------------------------------------------------------------
__global__ void gnn_zero_kernel(float* __restrict__ ws, long n) {
    long i = (long)blockIdx.x * blockDim.x + threadIdx.x;
    if (i < n) ws[i] = 0.0f;
}

// ---------------------------------------------------------------------------
// Kernel 2: edge gather + scatter-add (mean-aggregation numerator + degree)
// 32 lanes per edge; each lane gathers float4 (4 channels) from the source
// embedding row and does 4 native fp32 global atomics into agg[dst].
// Working set fits in 192MB L2, so this is an L2-atomic-bound stage.
// ---------------------------------------------------------------------------
__global__ void gnn_edge_scatter_kernel(const int* __restrict__ node_index,
                                        const int* __restrict__ edge_index,
                                        const float* __restrict__ emb,
                                        float* __restrict__ agg,
                                        float* __restrict__ deg) {
    long gid = (long)blockIdx.x * blockDim.x + threadIdx.x;
    long e   = gid >> 5;          // edge id
    int  c   = (int)(gid & 31);   // channel group (4 channels per lane)
    if (e >= N_EDGES) return;

    int s  = edge_index[e];            // src row of x
    int d  = edge_index[N_EDGES + e];  // dst row of agg
    int xi = node_index[s];            // x = emb_weight[node_index] -> gather row

    const float4 v = *(const float4*)(emb + (size_t)xi * HID + c * 4);
    float* ap = agg + (size_t)d * HID + c * 4;
#if defined(__HIP_DEVICE_COMPILE__)
    unsafeAtomicAdd(ap + 0, v.x);   // global_atomic_add_f32 (hardware FP atomic)
    unsafeAtomicAdd(ap + 1, v.y);
    unsafeAtomicAdd(ap + 2, v.z);
    unsafeAtomicAdd(ap + 3, v.w);
    if (c == 0) unsafeAtomicAdd(deg + d, 1.0f);
#endif
}

// ---------------------------------------------------------------------------
// Kernel 3: fused  h = relu((agg/deg)@Wmsg^T + nf@Wnode^T + bg)
//                  p = h@Wpred^T + bp ;  out = log_softmax(p)
// One block = 16 nodes, 128 threads (4 waves). Full fp32 via
// V_WMMA_F32_16X16X4_F32 (matches fp32 reference numerics).
//
// Fragment layout for 16x16x4 f32 (ISA 7.12.2):
//   A lane l, reg r : A[l%16][k0 + 2*(l/16) + r]      -> float2 row-major load
//   B lane l, reg r : B[k0 + 2*(l/16) + r][l%16]      -> since GEMM is X@W^T,
//                     B[k][j] = W[j][k]               -> float2 load of W row
//   D lane l, reg r : D[r + 8*(l/16)][l%16]
// ---------------------------------------------------------------------------
__launch_bounds__(128)
__global__ void gnn_fused_gemm_kernel(const float* __restrict__ agg,
                                      const float* __restrict__ deg,
                                      const float* __restrict__ nf,
                                      const float* __restrict__ Wmsg,
                                      const float* __restrict__ Wnode,
                                      const float* __restrict__ bg,
                                      const float* __restrict__ Wpred,
                                      const float* __restrict__ bp,
                                      float* __restrict__ out) {
    __shared__ float lds_h[16 * HID];   // relu'd hidden tile, 8 KB
    __shared__ float lds_p[16 * 48];    // pred tile (40 padded to 48), 3 KB

    const int tid   = threadIdx.x;
    const int lane  = tid & 31;
    const int wave  = tid >> 5;
    const int mrow  = lane & 15;           // tile row for A / tile col for D
    const int khalf = (lane >> 4) << 1;    // +0 or +2 k sub-offset
    const int mbase = blockIdx.x * 16;
    const int row   = mbase + mrow;        // N_NODES % 16 == 0 -> always valid

    const float invd = 1.0f / fmaxf(deg[row], 1.0f);   // mean divisor (clip>=1)

    const float* aggRow = agg + (size_t)row * HID;
    const float* nfRow  = nf  + (size_t)row * HID;

    // Each wave owns two 16-wide column tiles of h: jb0 = 16w, jb1 = 16(w+4)
    const int jb0 = 16 * wave;
    const int jb1 = 16 * (wave + 4);
    const float* wm0 = Wmsg  + (size_t)(jb0 + mrow) * HID;
    const float* wm1 = Wmsg  + (size_t)(jb1 + mrow) * HID;
    const float* wn0 = Wnode + (size_t)(jb0 + mrow) * HID;
    const float* wn1 = Wnode + (size_t)(jb1 + mrow) * HID;

    v8f acc0 = {};
    v8f acc1 = {};

    #pragma unroll 4
    for (int k0 = 0; k0 < HID; k0 += 4) {
        const int ko = k0 + khalf;
        v2f a1 = *(const v2f*)(aggRow + ko);
        a1.x *= invd; a1.y *= invd;                 // scale by 1/deg (mean)
        const v2f a2 = *(const v2f*)(nfRow + ko);

        v2f b;
        b = *(const v2f*)(wm0 + ko);
        acc0 = __builtin_amdgcn_wmma_f32_16x16x4_f32(false, a1, false, b,
                                                     (short)0, acc0, false, false);
        b = *(const v2f*)(wn0 + ko);
        acc0 = __builtin_amdgcn_wmma_f32_16x16x4_f32(false, a2, false, b,
                                                     (short)0, acc0, false, false);
        b = *(const v2f*)(wm1 + ko);
        acc1 = __builtin_amdgcn_wmma_f32_16x16x4_f32(false, a1, false, b,
                                                     (short)0, acc1, false, false);
        b = *(const v2f*)(wn1 + ko);
        acc1 = __builtin_amdgcn_wmma_f32_16x16x4_f32(false, a2, false, b,
                                                     (short)0, acc1, false, false);
    }

    // bias + relu, scatter D tiles into LDS (column index of D == lane%16)
    const float bias0 = bg[jb0 + mrow];
    const float bias1 = bg[jb1 + mrow];
    const int   rbase = (lane >> 4) << 3;   // D rows r..r+7 live at +0 or +8
    #pragma unroll
    for (int r = 0; r < 8; ++r) {
        const int dr = r + rbase;
        lds_h[dr * HID + jb0 + mrow] = fmaxf(acc0[r] + bias0, 0.0f);
        lds_h[dr * HID + jb1 + mrow] = fmaxf(acc1[r] + bias1, 0.0f);
    }
    __syncthreads();

    // ---- prediction GEMM: 3 column tiles cover PRED=40 (<=48). All 4 waves
    // run the WMMA loop (EXEC all-ones); wave 3 computes a throwaway copy of
    // tile 0 and skips the store.
    const int jt   = (wave < 3) ? wave : 0;
    const int jbp  = 16 * jt;
    const int jrow = jbp + mrow;
    const int jc   = (jrow < PRED) ? jrow : (PRED - 1);   // clamp pad columns
    const float* wp = Wpred + (size_t)jc * HID;

    v8f accp = {};
    #pragma unroll 4
    for (int k0 = 0; k0 < HID; k0 += 4) {
        const int ko = k0 + khalf;
        const v2f a = *(const v2f*)(&lds_h[mrow * HID + ko]);   // ds_load_b64
        const v2f b = *(const v2f*)(wp + ko);
        accp = __builtin_amdgcn_wmma_f32_16x16x4_f32(false, a, false, b,
                                                     (short)0, accp, false, false);
    }
    if (wave < 3) {
        const float bpv = bp[jc];
        #pragma unroll
        for (int r = 0; r < 8; ++r) {
            lds_p[(r + rbase) * 48 + jbp + mrow] = accp[r] + bpv;
        }
    }
    __syncthreads();

    // ---- log_softmax over the 40 valid columns, one thread per node row
    if (tid < 16) {
        const int node = mbase + tid;
        if (node < N_NODES) {
            const float* pr = &lds_p[tid * 48];
            float m = -3.402823466e38f;
            #pragma unroll
            for (int c = 0; c < PRED; ++c) m = fmaxf(m, pr[c]);
            float s = 0.0f;
            #pragma unroll
            for (int c = 0; c < PRED; ++c) s += __expf(pr[c] - m);
            const float lse = m + __logf(s);
            float* po = out + (size_t)node * PRED;
            #pragma unroll
            for (int c = 0; c < PRED; ++c) po[c] = pr[c] - lse;
        }
    }
}

// ---------------------------------------------------------------------------
extern "C" void kernel_launch(void* const* d_in, const int* in_sizes, int n_in,
                              void* d_out, int out_size, void* d_ws, size_t ws_size,
                              hipStream_t stream) {
    const int*   node_index   = (const int*)d_in[0];
    const float* node_feature = (const float*)d_in[1];
    const int*   edge_index   = (const int*)d_in[2];
    const float* emb_weight   = (const float*)d_in[3];
    const float* W_msg        = (const float*)d_in[4];
    const float* W_node       = (const float*)d_in[5];
    const float* b_graph      = (const float*)d_in[6];
    const float* W_pred       = (const float*)d_in[7];
    const float* b_pred       = (const float*)d_in[8];
    float*       out          = (float*)d_out;

    // workspace carve: agg[N][HID] then deg[N]   (~51.6 MB)
    float* agg = (float*)d_ws;
    float* deg = agg + (size_t)N_NODES * HID;

    const long nz = (long)N_NODES * HID + N_NODES;
    gnn_zero_kernel<<<(int)((nz + 255) / 256), 256, 0, stream>>>(agg, nz);

    const long nth = (long)N_EDGES * 32;   // 32 lanes per edge
    gnn_edge_scatter_kernel<<<(int)((nth + 255) / 256), 256, 0, stream>>>(
        node_index, edge_index, emb_weight, agg, deg);

    gnn_fused_gemm_kernel<<<(N_NODES + 15) / 16, 128, 0, stream>>>(
        agg, deg, node_feature, W_msg, W_node, b_graph, W_pred, b_pred, out);
}